// LSTMClassifier_80041010528613
// MI455X (gfx1250) — compile-verified
//
#include <hip/hip_runtime.h>

typedef __attribute__((ext_vector_type(2))) float v2f;
typedef __attribute__((ext_vector_type(8))) float v8f;

#define NB     32      // batch
#define TSTEPS 3       // only t=0,1,2 affect the output slice out[:,2:3,:]
#define TFULL  4096
#define FIN    11
#define DIN    124     // lstm input features
#define DINP   128     // padded K
#define H      128
#define Z4     512     // 4*H

__device__ __forceinline__ float leaky(float x) { return x >= 0.0f ? x : 0.2f * x; }
__device__ __forceinline__ float sigm(float x)  { return 1.0f / (1.0f + expf(-x)); }

// LDS layout (floats)
#define OFF_IN  0                             // 3*32*128 = 12288
#define OFF_H   (OFF_IN + TSTEPS * NB * DINP) // 4096
#define OFF_C   (OFF_H + NB * H)              // 4096
#define OFF_Z   (OFF_C + NB * H)              // 32*512 = 16384
#define SMEM_FLOATS (OFF_Z + NB * Z4)         // 36864 floats = 144 KB (WGP has 320 KB)

__global__ __launch_bounds__(256)
void lstm_head_fused(
    const float* __restrict__ data,
    const float* __restrict__ W_M1, const float* __restrict__ b_M1,
    const float* __restrict__ W_M2, const float* __restrict__ b_M2,
    const float* __restrict__ W_T1, const float* __restrict__ b_T1,
    const float* __restrict__ W_T2, const float* __restrict__ b_T2,
    const float* __restrict__ W_S1, const float* __restrict__ b_S1,
    const float* __restrict__ W_S2, const float* __restrict__ b_S2,
    const float* __restrict__ Wk,  const float* __restrict__ Wr, const float* __restrict__ b_l,
    const float* __restrict__ W_P1, const float* __restrict__ b_P1,
    const float* __restrict__ W_P2, const float* __restrict__ b_P2,
    const float* __restrict__ W_P3, const float* __restrict__ b_P3,
    const float* __restrict__ W_P4, const float* __restrict__ b_P4,
    const float* __restrict__ W_P5, const float* __restrict__ b_P5,
    float* __restrict__ out)
{
    extern __shared__ float sm[];
    const int tid = threadIdx.x;

    // zero h, c
    for (int i = tid; i < NB * H; i += 256) { sm[OFF_H + i] = 0.0f; sm[OFF_C + i] = 0.0f; }

    // ---------------- Stage A: per-(b,t) feature MLPs -> lstm_in rows in LDS -------------
    if (tid < NB * TSTEPS) {
        const int t = tid / NB, b = tid % NB;
        const float* xr = data + ((size_t)b * TFULL + t) * FIN;
        float x[FIN];
        #pragma unroll
        for (int i = 0; i < FIN; ++i) x[i] = xr[i];

        float* row = &sm[OFF_IN + (t * NB + b) * DINP];

        // M = leaky(leaky(body@W_M1+b)@W_M2+b), body = x[4:10]
        float m1[8];
        #pragma unroll
        for (int o = 0; o < 8; ++o) {
            float s = b_M1[o];
            #pragma unroll
            for (int i = 0; i < 6; ++i) s += x[4 + i] * W_M1[i * 8 + o];
            m1[o] = leaky(s);
        }
        #pragma unroll
        for (int o = 0; o < 16; ++o) {
            float s = b_M2[o];
            #pragma unroll
            for (int i = 0; i < 8; ++i) s += m1[i] * W_M2[i * 16 + o];
            row[o] = leaky(s);                       // M -> row[0:16]
        }
        // Tsk = |leaky(x@W_T1+b)| ; Tsk = |leaky(Tsk@W_T2+b)| ; Psk = log1p(Tsk)
        float t1[8];
        #pragma unroll
        for (int o = 0; o < 8; ++o) {
            float s = b_T1[o];
            #pragma unroll
            for (int i = 0; i < FIN; ++i) s += x[i] * W_T1[i * 8 + o];
            t1[o] = fabsf(leaky(s));
        }
        #pragma unroll
        for (int o = 0; o < 16; ++o) {
            float s = b_T2[o];
            #pragma unroll
            for (int i = 0; i < 8; ++i) s += t1[i] * W_T2[i * 16 + o];
            float v = fabsf(leaky(s));
            row[16 + o] = v;                         // Tsk -> row[16:32]
            row[32 + o] = log1pf(v);                 // Psk -> row[32:48]
        }
        row[48] = log1pf(x[1]);                      // Pa

        // S = dense(dense(s_in, W_S1), W_S2); s_in = [M,Tsk,Psk,Pa, env=x[1:4], body=x[4:10]]
        float s1[32];
        #pragma unroll
        for (int o = 0; o < 32; ++o) {
            float s = b_S1[o];
            #pragma unroll
            for (int i = 0; i < 49; ++i) s += row[i] * W_S1[i * 32 + o];
            #pragma unroll
            for (int i = 0; i < 3; ++i)  s += x[1 + i] * W_S1[(49 + i) * 32 + o];
            #pragma unroll
            for (int i = 0; i < 6; ++i)  s += x[4 + i] * W_S1[(52 + i) * 32 + o];
            s1[o] = leaky(s);
        }
        #pragma unroll
        for (int o = 0; o < 64; ++o) {
            float s = b_S2[o];
            #pragma unroll
            for (int i = 0; i < 32; ++i) s += s1[i] * W_S2[i * 64 + o];
            row[49 + o] = leaky(s);                  // S -> row[49:113]
        }
        #pragma unroll
        for (int i = 0; i < FIN; ++i) row[113 + i] = x[i];   // data -> row[113:124]
        #pragma unroll
        for (int i = DIN; i < DINP; ++i) row[i] = 0.0f;      // zero-pad K
    }

    // ---------------- Stage B: 3 LSTM steps, z = in@Wk + b_l + h@Wr via fp32 WMMA --------
    const int wave = tid >> 5;
    const int lane = tid & 31;
    const bool lo  = lane < 16;
    const int l15  = lane & 15;

    for (int t = 0; t < TSTEPS; ++t) {
        __syncthreads();
        const float* inRow = &sm[OFF_IN + t * NB * DINP];

        for (int tt = 0; tt < 8; ++tt) {          // 8 waves x 8 tiles = 64 C-tiles (2x32)
            const int tile = wave * 8 + tt;
            const int mt = tile >> 5, nt = tile & 31;
            const int col = nt * 16 + l15;        // N index (B/C/D lane col)
            const int m   = mt * 16 + l15;        // M index (A lane row)
            const float bias = b_l[col];
            v8f acc;
            #pragma unroll
            for (int r = 0; r < 8; ++r) acc[r] = bias;

            // in(32x128-padded) @ Wk(124x512)
            #pragma unroll 8
            for (int kt = 0; kt < 32; ++kt) {
                const int k0 = kt * 4 + (lo ? 0 : 2);
                v2f a, bm;
                a.x  = inRow[m * DINP + k0];
                a.y  = inRow[m * DINP + k0 + 1];
                bm.x = (k0     < DIN) ? Wk[k0 * Z4 + col]       : 0.0f;
                bm.y = (k0 + 1 < DIN) ? Wk[(k0 + 1) * Z4 + col] : 0.0f;
                acc = __builtin_amdgcn_wmma_f32_16x16x4_f32(false, a, false, bm,
                                                            (short)0, acc, false, false);
            }
            // h(32x128) @ Wr(128x512)
            #pragma unroll 8
            for (int kt = 0; kt < 32; ++kt) {
                const int k0 = kt * 4 + (lo ? 0 : 2);
                v2f a, bm;
                a.x  = sm[OFF_H + m * H + k0];
                a.y  = sm[OFF_H + m * H + k0 + 1];
                bm.x = Wr[k0 * Z4 + col];
                bm.y = Wr[(k0 + 1) * Z4 + col];
                acc = __builtin_amdgcn_wmma_f32_16x16x4_f32(false, a, false, bm,
                                                            (short)0, acc, false, false);
            }
            #pragma unroll
            for (int r = 0; r < 8; ++r) {
                const int rowD = mt * 16 + r + (lo ? 0 : 8);
                sm[OFF_Z + rowD * Z4 + col] = acc[r];
            }
        }
        __syncthreads();

        // gate update: i,f,g,o split of z
        for (int i = tid; i < NB * H; i += 256) {
            const int b = i >> 7, j = i & 127;
            const float iz = sm[OFF_Z + b * Z4 + j];
            const float fz = sm[OFF_Z + b * Z4 + 128 + j];
            const float gz = sm[OFF_Z + b * Z4 + 256 + j];
            const float oz = sm[OFF_Z + b * Z4 + 384 + j];
            const float cc = sigm(fz) * sm[OFF_C + i] + sigm(iz) * leaky(gz);
            sm[OFF_C + i] = cc;
            sm[OFF_H + i] = sigm(oz) * leaky(cc);
        }
    }
    __syncthreads();

    // ---------------- Stage C: head on h[t=2], softmax, outputs --------------------------
    if (tid < NB) {
        const int b = tid;
        const float* hb = &sm[OFF_H + b * H];
        float d1[64], d2[32], d3[16], d4[8], d5[3];
        #pragma unroll 8
        for (int o = 0; o < 64; ++o) {
            float s = b_P1[o];
            for (int i = 0; i < 128; ++i) s += hb[i] * W_P1[i * 64 + o];
            d1[o] = leaky(s);
        }
        #pragma unroll 8
        for (int o = 0; o < 32; ++o) {
            float s = b_P2[o];
            for (int i = 0; i < 64; ++i) s += d1[i] * W_P2[i * 32 + o];
            d2[o] = leaky(s);
        }
        #pragma unroll
        for (int o = 0; o < 16; ++o) {
            float s = b_P3[o];
            for (int i = 0; i < 32; ++i) s += d2[i] * W_P3[i * 16 + o];
            d3[o] = leaky(s);
        }
        #pragma unroll
        for (int o = 0; o < 8; ++o) {
            float s = b_P4[o];
            for (int i = 0; i < 16; ++i) s += d3[i] * W_P4[i * 8 + o];
            d4[o] = leaky(s);
        }
        #pragma unroll
        for (int o = 0; o < 3; ++o) {
            float s = b_P5[o];
            for (int i = 0; i < 8; ++i) s += d4[i] * W_P5[i * 3 + o];
            d5[o] = leaky(s);
        }
        const float mx = fmaxf(d5[0], fmaxf(d5[1], d5[2]));
        float e0 = expf(d5[0] - mx), e1 = expf(d5[1] - mx), e2 = expf(d5[2] - mx);
        const float inv = 1.0f / (e0 + e1 + e2);
        e0 *= inv; e1 *= inv; e2 *= inv;

        // output (32,1,3)
        out[b * 3 + 0] = e0; out[b * 3 + 1] = e1; out[b * 3 + 2] = e2;
        // x = concat(data[:,2,1:11], output) -> (32,1,13), appended after output (96 floats)
        const float* xr2 = data + ((size_t)b * TFULL + 2) * FIN;
        float* xo = out + NB * 3 + b * 13;
        #pragma unroll
        for (int j = 0; j < 10; ++j) xo[j] = xr2[1 + j];
        xo[10] = e0; xo[11] = e1; xo[12] = e2;
    }
}

extern "C" void kernel_launch(void* const* d_in, const int* in_sizes, int n_in,
                              void* d_out, int out_size, void* d_ws, size_t ws_size,
                              hipStream_t stream) {
    (void)in_sizes; (void)n_in; (void)d_ws; (void)ws_size; (void)out_size;
    const float* P[26];
    for (int i = 0; i < 26; ++i) P[i] = (const float*)d_in[i];
    const size_t smem = (size_t)SMEM_FLOATS * sizeof(float); // 144 KB dynamic LDS
    hipFuncSetAttribute((const void*)lstm_head_fused,
                        hipFuncAttributeMaxDynamicSharedMemorySize, (int)smem);
    lstm_head_fused<<<1, 256, smem, stream>>>(
        P[0],
        P[1],  P[2],  P[3],  P[4],
        P[5],  P[6],  P[7],  P[8],
        P[9],  P[10], P[11], P[12],
        P[13], P[14], P[15],
        P[16], P[17], P[18], P[19],
        P[20], P[21], P[22], P[23],
        P[24], P[25],
        (float*)d_out);
}